// PlasticityController_79542794322002
// MI455X (gfx1250) — compile-verified
//
#include <hip/hip_runtime.h>

typedef float v2f __attribute__((ext_vector_type(2)));
typedef float v8f __attribute__((ext_vector_type(8)));

// Problem constants (fixed by the reference)
constexpr int   T_STEPS = 128;
constexpr int   B       = 8;
constexpr int   NB      = 256;
constexpr int   P       = 16;
constexpr int   N       = NB * P;     // 4096
constexpr int   E       = 4096;
constexpr float SYN_LR   = 0.01f;
constexpr float MAX_NORM = 1.0f;
constexpr float DMAX     = 0.1f;
constexpr float EPS      = 1e-8f;

// One wave32 per edge. W tile kept in registers in the WMMA C/D layout:
//   VGPR v, lane l: element (M = v + 8*(l>=16), N = l&15)
// hebb = A @ Bm with A[i,b] = post[b,i]*iv[b]/B (16x8), Bm[b,j] = pre[b,j] (8x16),
// computed as two V_WMMA_F32_16X16X4_F32 accumulating into the same C.
// Per step there is exactly ONE 5-level shuffle-reduction tree (3 sums fused)
// plus one batched bpermute round-trip for the post2 redistribution.
__global__ __launch_bounds__(256) void plasticity_scan_kernel(
    const float* __restrict__ states,   // (T,B,N)
    const float* __restrict__ elig,     // (T,B,N)
    const float* __restrict__ inv,      // (T,B,1)
    const float* __restrict__ var,      // (T,B,N)
    const float* __restrict__ Winit,    // (E,P,P)
    const int*   __restrict__ rows,     // (E,)
    const int*   __restrict__ cols,     // (E,)
    const unsigned char* __restrict__ active, // (E,) bool
    float* __restrict__ Wout)           // (E,P,P)
{
  const int lane = threadIdx.x & 31;
  const int wave = threadIdx.x >> 5;
  const int e    = blockIdx.x * 8 + wave;
  const int x    = lane & 15;   // M for A, N for B/C/D
  const int h    = lane >> 4;   // wave half

  const int   row  = rows[e];
  const int   col  = cols[e];
  const float mask = active[e] ? 1.0f : 0.0f;

  const int rbase = row * P + x;   // post gather column
  const int cbase = col * P + x;   // pre  gather column

  // b indices this lane owns: k -> chunk c = k>>1, A/B vgpr v = k&1, b = 4c + 2h + v
  int bidx[4];
  #pragma unroll
  for (int k = 0; k < 4; ++k) bidx[k] = 4 * (k >> 1) + 2 * h + (k & 1);

  // Loop-invariant shuffle source lanes for post2 redistribution:
  // lane needs post2[M = v + 8h]; partial[x] lives on lanes x and x+16.
  int srcLo[8], srcHi[8];
  #pragma unroll
  for (int v = 0; v < 8; ++v) { srcLo[v] = v + 8 * h; srcHi[v] = v + 8 * h + 16; }

  // Load W in C/D layout
  v8f W;
  #pragma unroll
  for (int v = 0; v < 8; ++v)
    W[v] = Winit[(size_t)e * (P * P) + (size_t)(v + 8 * h) * P + x];

  for (int t = 0; t < T_STEPS; ++t) {
    float a[4], bm[4];
    float partial = 0.0f;
    #pragma unroll
    for (int k = 0; k < 4; ++k) {
      const int    b   = bidx[k];
      const size_t off = ((size_t)t * B + b) * (size_t)N;
      const float  s   = states[off + rbase];
      const float  vv  = var[off + rbase];
      const float  post = vv * (1.0f - s * s);            // mod
      const float  ivs  = inv[t * B + b] * (1.0f / B);
      a[k]  = post * ivs;                                 // A element (scaled)
      partial += a[k] * post;                             // post^2 * iv / B
      bm[k] = elig[off + cbase];                          // B element
    }

    // Prefetch next step's gather lines (global_prefetch_b8)
    if (t + 1 < T_STEPS) {
      const size_t offn = ((size_t)(t + 1) * B + bidx[0]) * (size_t)N;
      __builtin_prefetch(&states[offn + rbase], 0, 0);
      __builtin_prefetch(&var[offn + rbase],    0, 0);
      __builtin_prefetch(&elig[offn + cbase],   0, 0);
    }

    // hebb = 16x8 @ 8x16 via two K=4 f32 WMMAs, accumulated in C
    v8f hebb = {};
    v2f A0 = { a[0],  a[1]  }, B0 = { bm[0], bm[1] };
    v2f A1 = { a[2],  a[3]  }, B1 = { bm[2], bm[3] };
    hebb = __builtin_amdgcn_wmma_f32_16x16x4_f32(false, A0, false, B0,
                                                 (short)0, hebb, false, false);
    hebb = __builtin_amdgcn_wmma_f32_16x16x4_f32(false, A1, false, B1,
                                                 (short)0, hebb, false, false);

    // post2[M = v+8h] per vgpr: fetch both halves directly (one bpermute batch)
    float p2[8];
    #pragma unroll
    for (int v = 0; v < 8; ++v)
      p2[v] = __shfl(partial, srcLo[v], 32) + __shfl(partial, srcHi[v], 32);

    // dw = hebb - post2[M] * W ; accumulate the three tile sums per lane
    v8f   dw;
    float sdd = 0.0f;   // sum dw*dw
    float swd = 0.0f;   // sum W*dw
    float sww = 0.0f;   // sum W*W
    #pragma unroll
    for (int v = 0; v < 8; ++v) {
      const float d = hebb[v] - p2[v] * W[v];
      dw[v] = d;
      sdd  += d * d;
      swd  += W[v] * d;
      sww  += W[v] * W[v];
    }

    // Single fused 5-level reduction tree for all three sums
    #pragma unroll
    for (int m = 1; m < 32; m <<= 1) {
      sdd += __shfl_xor(sdd, m, 32);
      swd += __shfl_xor(swd, m, 32);
      sww += __shfl_xor(sww, m, 32);
    }

    const float dn   = sqrtf(sdd);
    const float sc   = fminf(1.0f, DMAX / (dn + EPS)) * (SYN_LR * mask);
    // ||W + sc*dw||^2 without a second reduction tree
    const float wn2  = sww + 2.0f * sc * swd + sc * sc * sdd;
    const float wsc  = fminf(1.0f, MAX_NORM / (sqrtf(wn2) + EPS));

    #pragma unroll
    for (int v = 0; v < 8; ++v)
      W[v] = (W[v] + sc * dw[v]) * wsc;
  }

  #pragma unroll
  for (int v = 0; v < 8; ++v)
    Wout[(size_t)e * (P * P) + (size_t)(v + 8 * h) * P + x] = W[v];
}

extern "C" void kernel_launch(void* const* d_in, const int* in_sizes, int n_in,
                              void* d_out, int out_size, void* d_ws, size_t ws_size,
                              hipStream_t stream) {
  // setup_inputs order:
  // 0 states, 1 elig, 2 act (unused for W), 3 inv, 4 var, 5 weight_values,
  // 6 activity_bias (dead), 7 trophic_map (dead), 8 weight_rows, 9 weight_cols,
  // 10 active_blocks
  const float* states = (const float*)d_in[0];
  const float* elig   = (const float*)d_in[1];
  const float* inv    = (const float*)d_in[3];
  const float* var    = (const float*)d_in[4];
  const float* Winit  = (const float*)d_in[5];
  const int*   rows   = (const int*)d_in[8];
  const int*   cols   = (const int*)d_in[9];
  const unsigned char* active = (const unsigned char*)d_in[10];
  float* Wout = (float*)d_out;

  dim3 block(256);            // 8 waves per block
  dim3 grid(E / 8);           // 4096 waves total, one per edge
  hipLaunchKernelGGL(plasticity_scan_kernel, grid, block, 0, stream,
                     states, elig, inv, var, Winit, rows, cols, active, Wout);
}